// SparseVoxelPixelLayer_85126251807208
// MI455X (gfx1250) — compile-verified
//
#include <hip/hip_runtime.h>
#include <stdint.h>

// Problem constants (match the JAX reference)
#define NUM_VOXELS 524288
#define NUM_PIXELS 262144
#define BATCH      4

#define TILE    2048      // edges staged per LDS tile
#define THREADS 256       // 8 wave32s per block
#define MAXBLK  2048      // persistent blocks (grid-stride over tiles)

// -------------------------------------------------------------------------
// Zero-init the output (harness poisons d_out to 0xAA before timing).
// -------------------------------------------------------------------------
__global__ void zero_out_kernel(float* __restrict__ out, int n) {
    int n4 = n >> 2;
    float4* out4 = reinterpret_cast<float4*>(out);
    for (int i = blockIdx.x * blockDim.x + threadIdx.x; i < n4;
         i += gridDim.x * blockDim.x) {
        out4[i] = make_float4(0.f, 0.f, 0.f, 0.f);
    }
    for (int i = (n4 << 2) + blockIdx.x * blockDim.x + threadIdx.x; i < n;
         i += gridDim.x * blockDim.x) {
        out[i] = 0.f;
    }
}

// -------------------------------------------------------------------------
// gfx1250 async global->LDS copy (ASYNCcnt-tracked). Low 32 bits of a flat
// pointer to __shared__ are the LDS byte offset (aperture lives in the
// upper 32 bits), which is what the VDST lane operand expects.
// The edge stream is ~200 MB (> 192 MB L2), so mark it non-temporal to
// keep x (8 MB) and out (4 MB) resident in L2 for the gather/scatter.
// -------------------------------------------------------------------------
__device__ __forceinline__ uint32_t lds_off32(const void* p) {
    return (uint32_t)(uintptr_t)p;
}

__device__ __forceinline__ void async_copy_b128_nt(uint32_t lds_byte_off,
                                                   const void* gaddr) {
    asm volatile("global_load_async_to_lds_b128 %0, %1, off th:TH_LOAD_NT"
                 :
                 : "v"(lds_byte_off),
                   "v"((unsigned long long)(uintptr_t)gaddr)
                 : "memory");
}

// Gather 4 batch values of x (L2-resident) and scatter-add into out.
__device__ __forceinline__ void scatter_edge(const float* __restrict__ x,
                                             float* __restrict__ out,
                                             float w, int v, int p) {
    w = fminf(fmaxf(w, -1.0f), 1.0f);   // clip(edge_weights, -1, 1)
#pragma unroll
    for (int b = 0; b < BATCH; ++b) {
        const float xv = x[(size_t)b * NUM_VOXELS + v];
        unsafeAtomicAdd(out + (size_t)b * NUM_PIXELS + p, w * xv);
    }
}

// -------------------------------------------------------------------------
// Edge scatter kernel: double-buffered async staging of the edge stream,
// L2-resident gathers of x, fp32 atomic scatter into out.
// -------------------------------------------------------------------------
__global__ void __launch_bounds__(THREADS)
sparse_voxel_pixel_kernel(const float* __restrict__ x,     // [B, NUM_VOXELS]
                          const float* __restrict__ ew,    // [E]
                          const int*   __restrict__ vidx,  // [E]
                          const int*   __restrict__ pidx,  // [E]
                          float*       __restrict__ out,   // [B, NUM_PIXELS]
                          int nedges) {
    __shared__ __align__(16) float s_w [2][TILE];
    __shared__ __align__(16) int   s_vi[2][TILE];
    __shared__ __align__(16) int   s_pi[2][TILE];

    const int tid    = threadIdx.x;
    const int ntiles = nedges / TILE;

    // Issue one tile's async copies: 3 arrays * 2 b128 chunks per thread.
    // Per wave that is 6 ASYNCcnt-tracked instructions.
    auto issue_tile = [&](int t, int buf) {
        const int base = t * TILE;
#pragma unroll
        for (int j = 0; j < 2; ++j) {
            const int c = (j * THREADS + tid) * 4;   // element idx in tile
            async_copy_b128_nt(lds_off32(&s_w [buf][c]), (const void*)(ew   + base + c));
            async_copy_b128_nt(lds_off32(&s_vi[buf][c]), (const void*)(vidx + base + c));
            async_copy_b128_nt(lds_off32(&s_pi[buf][c]), (const void*)(pidx + base + c));
        }
    };

    int t = blockIdx.x;
    if (t < ntiles) {
        int buf = 0;
        issue_tile(t, buf);                       // prologue: fill buffer 0

        for (; t < ntiles; t += gridDim.x) {
            const int  tn       = t + gridDim.x;
            const bool has_next = tn < ntiles;
            if (has_next) {
                issue_tile(tn, buf ^ 1);          // overlap next tile's copy
                asm volatile("s_wait_asynccnt 6" ::: "memory");  // tile t done
            } else {
                asm volatile("s_wait_asynccnt 0" ::: "memory");
            }
            __syncthreads();                      // tile t visible to all waves

            // Consume the tile with b128 LDS reads (4 edges per ds_load).
#pragma unroll
            for (int k = 0; k < TILE / (THREADS * 4); ++k) {
                const int q = k * THREADS + tid;  // float4/int4 index in tile
                const float4 w4 = reinterpret_cast<const float4*>(s_w [buf])[q];
                const int4   v4 = reinterpret_cast<const int4*  >(s_vi[buf])[q];
                const int4   p4 = reinterpret_cast<const int4*  >(s_pi[buf])[q];
                scatter_edge(x, out, w4.x, v4.x, p4.x);
                scatter_edge(x, out, w4.y, v4.y, p4.y);
                scatter_edge(x, out, w4.z, v4.z, p4.z);
                scatter_edge(x, out, w4.w, v4.w, p4.w);
            }
            __syncthreads();                      // buf fully consumed
            buf ^= 1;
        }
    }

    // Tail: edges beyond the last full tile (none for E = 2^24).
    const long long tail = (long long)ntiles * TILE;
    for (long long e = tail + (long long)blockIdx.x * THREADS + tid;
         e < nedges; e += (long long)gridDim.x * THREADS) {
        scatter_edge(x, out, ew[e], vidx[e], pidx[e]);
    }
}

// -------------------------------------------------------------------------
// Launcher. Inputs (setup_inputs order): x, edge_weights, voxel_indices,
// pixel_indices. Integer inputs arrive as int32 per the harness contract.
// -------------------------------------------------------------------------
extern "C" void kernel_launch(void* const* d_in, const int* in_sizes, int n_in,
                              void* d_out, int out_size, void* d_ws, size_t ws_size,
                              hipStream_t stream) {
    const float* x    = (const float*)d_in[0];
    const float* ew   = (const float*)d_in[1];
    const int*   vidx = (const int*)d_in[2];
    const int*   pidx = (const int*)d_in[3];
    float*       out  = (float*)d_out;

    const int nedges = in_sizes[1];

    // 1) out = 0
    zero_out_kernel<<<1024, 256, 0, stream>>>(out, out_size);

    // 2) gather-multiply-scatter over the edge list
    int ntiles = nedges / TILE;
    int blocks = ntiles < MAXBLK ? (ntiles > 0 ? ntiles : 1) : MAXBLK;
    sparse_voxel_pixel_kernel<<<blocks, THREADS, 0, stream>>>(
        x, ew, vidx, pidx, out, nedges);
}